// GATModel_83459804496010
// MI455X (gfx1250) — compile-verified
//
#include <hip/hip_runtime.h>
#include <hip/hip_bf16.h>

#define BB 16
#define NN 166
#define FF 400
#define DHH 400
#define TOPM 5
#define SLOPE 0.2f
#define MROWS (BB * NN)          // 2656, exactly 166 tiles of 16
#define FPAD 401                 // odd LDS row stride -> conflict-free banks

typedef float v2f __attribute__((ext_vector_type(2)));
typedef float v8f __attribute__((ext_vector_type(8)));
typedef unsigned int u32x4 __attribute__((ext_vector_type(4)));
typedef int i32x4 __attribute__((ext_vector_type(4)));
typedef int i32x8 __attribute__((ext_vector_type(8)));

#define NEG_INF (-__builtin_inff())

__device__ __forceinline__ float leaky(float x) {
    return x >= 0.f ? x : SLOPE * x;
}

// ---------------------------------------------------------------------------
// TDM: DMA a 16-row x 400-col fp32 tile (row stride 400 floats) from global to
// LDS.  rows_avail = #valid rows at gsrc; the TDM zero-fills OOB rows for us.
// D# bit layout per CDNA5 ISA ch.8 (group0 128b, group1 256b, 2D tensor).
// This toolchain exposes the 6-arg builtin: (g0, g1, g2, g3, g4, cpol).
// ---------------------------------------------------------------------------
__device__ __forceinline__ void tdm_load_tile16x400(
    const float* gsrc, unsigned int lds_byte_addr, int rows_avail) {
    unsigned long long ga = (unsigned long long)(uintptr_t)gsrc;
    u32x4 g0;
    g0.x = 1u;                                           // count=1, user mode
    g0.y = lds_byte_addr;                                // lds_addr
    g0.z = (unsigned int)ga;                             // global_addr[31:0]
    g0.w = (unsigned int)((ga >> 32) & 0x01FFFFFFu) | (2u << 30); // [56:32]|type=2
    i32x8 g1;
    g1[0] = (int)(2u << 16);                 // data_size=4B; no pad/iterate/mask
    g1[1] = (int)(((unsigned)FF & 0xFFFFu) << 16);       // tensor_dim0 lo
    g1[2] = (int)(((unsigned)rows_avail & 0xFFFFu) << 16); // dim0 hi=0 | tensor_dim1 lo
    g1[3] = (int)(((unsigned)FF & 0xFFFFu) << 16);       // dim1 hi=0 | tile_dim0=400
    g1[4] = 16;                                          // tile_dim1=16, tile_dim2=0
    g1[5] = FF;                                          // tensor_dim0_stride = 400
    g1[6] = 0;                                           // stride hi | dim1_stride lo
    g1[7] = 0;
    i32x4 g2;
    g2[0] = 1;                                           // tensor_dim2 = 1 (benign)
    g2[1] = 0; g2[2] = 0; g2[3] = 0;
    i32x4 g3 = {0, 0, 0, 0};
    i32x8 g4 = {0, 0, 0, 0, 0, 0, 0, 0};
    __builtin_amdgcn_tensor_load_to_lds(g0, g1, g2, g3, g4, 0);
}

// ---------------------------------------------------------------------------
// Kernel 1: all four projections g[z] = h @ W[z] fused in one wave per tile.
// The A fragment (h) is loaded once per k-step and reused by 4 WMMAs -> 4x
// matrix-op density, 1/4 the A traffic, and no pointer-array (keeps loads on
// the GLOBAL path instead of FLAT).  grid = (166, 25), block = 32 (1 wave).
// ---------------------------------------------------------------------------
__global__ __launch_bounds__(32) void proj_wmma(
    const float* __restrict__ h,
    const float* __restrict__ W0, const float* __restrict__ W1,
    const float* __restrict__ W2, const float* __restrict__ W3,
    float* __restrict__ gbase) {
    const int m0 = blockIdx.x * 16;
    const int n0 = blockIdx.y * 16;
    const int lane = threadIdx.x;
    const int hi = lane >> 4;         // 0: K={0,1}, 1: K={2,3}
    const int l  = lane & 15;

    const float* __restrict__ hrow = h + (size_t)(m0 + l) * FF;
    v8f acc0 = {}, acc1 = {}, acc2 = {}, acc3 = {};
    for (int k = 0; k < FF; k += 4) {
        if ((k & 31) == 0)
            __builtin_prefetch(hrow + k + 128, 0, 1);   // global_prefetch_b8
        const int ka = k + hi * 2;
        v2f a = *(const v2f*)(hrow + ka);               // h[m0+l][ka..ka+1]
        const size_t o0 = (size_t)ka * DHH + n0 + l;
        const size_t o1 = o0 + DHH;
        v2f b0, b1, b2, b3;
        b0.x = W0[o0]; b0.y = W0[o1];
        b1.x = W1[o0]; b1.y = W1[o1];
        b2.x = W2[o0]; b2.y = W2[o1];
        b3.x = W3[o0]; b3.y = W3[o1];
        acc0 = __builtin_amdgcn_wmma_f32_16x16x4_f32(false, a, false, b0, (short)0, acc0, false, false);
        acc1 = __builtin_amdgcn_wmma_f32_16x16x4_f32(false, a, false, b1, (short)0, acc1, false, false);
        acc2 = __builtin_amdgcn_wmma_f32_16x16x4_f32(false, a, false, b2, (short)0, acc2, false, false);
        acc3 = __builtin_amdgcn_wmma_f32_16x16x4_f32(false, a, false, b3, (short)0, acc3, false, false);
    }
    const size_t base = (size_t)(m0 + hi * 8) * DHH + n0 + l;
#pragma unroll
    for (int v = 0; v < 8; ++v) {
        const size_t ro = base + (size_t)v * DHH;
        gbase[0 * (size_t)MROWS * DHH + ro] = acc0[v];
        gbase[1 * (size_t)MROWS * DHH + ro] = acc1[v];
        gbase[2 * (size_t)MROWS * DHH + ro] = acc2[v];
        gbase[3 * (size_t)MROWS * DHH + ro] = acc3[v];
    }
}

// ---------------------------------------------------------------------------
// Kernel 2: pairwise scores for one branch.
// e[b,i,j] = adj[b,i,j] > 0 ? (a . leaky(gl[b,j,:] + gr[b,i,:]) + bias) : -inf
// gr tile staged by the Tensor Data Mover (contiguous, broadcast-read -> no
// bank conflicts); gl tile staged by VALU loads at odd stride 401 (16-way
// indexed reads, conflict-free).  grid = (11, 11, 16), block = 256.
// ---------------------------------------------------------------------------
__global__ __launch_bounds__(256) void scores_kernel(
    const float* __restrict__ gl, const float* __restrict__ gr,
    const float* __restrict__ avec, const float* __restrict__ adj,
    const float* __restrict__ bias_ptr, int has_bias,
    float* __restrict__ e) {
    __shared__ float s_gl[16 * FPAD];
    __shared__ float s_gr[16 * FF];
    __shared__ float s_a[FF];

    const int b  = blockIdx.z;
    const int j0 = blockIdx.x * 16;
    const int i0 = blockIdx.y * 16;
    const int tid = threadIdx.x;

    // wave 0 kicks off the TDM for the gr tile, waits TENSORcnt, then barrier
    if (tid < 32) {
        const float* gsrc = gr + ((size_t)b * NN + i0) * FF;
        tdm_load_tile16x400(gsrc, (unsigned int)(uintptr_t)&s_gr[0], NN - i0);
        __builtin_amdgcn_s_wait_tensorcnt(0);
    }

    for (int d = tid; d < FF; d += 256) s_a[d] = avec[d];
    for (int idx = tid; idx < 16 * FF; idx += 256) {
        const int r = idx / FF, d = idx - r * FF;
        const int j = j0 + r;
        s_gl[r * FPAD + d] = (j < NN) ? gl[((size_t)b * NN + j) * FF + d] : 0.f;
    }
    __syncthreads();

    const int ti = tid >> 4, tj = tid & 15;
    const int i = i0 + ti, j = j0 + tj;
    const float* __restrict__ pl = s_gl + tj * FPAD;
    const float* __restrict__ pr = s_gr + ti * FF;
    float acc = 0.f;
#pragma unroll 4
    for (int d = 0; d < FF; ++d) {
        const float t = pl[d] + pr[d];
        acc = fmaf(s_a[d], leaky(t), acc);
    }
    if (i < NN && j < NN) {
        const float av = adj[((size_t)b * NN + i) * NN + j];
        const float val = acc + (has_bias ? bias_ptr[0] : 0.f);
        e[((size_t)b * NN + i) * NN + j] = (av > 0.f) ? val : NEG_INF;
    }
}

// ---------------------------------------------------------------------------
// Kernel 3: in-place row softmax over j (length 166) for the global branch.
// ---------------------------------------------------------------------------
__global__ __launch_bounds__(256) void softmax_rows(float* __restrict__ e) {
    float* __restrict__ p = e + (size_t)blockIdx.x * NN;
    const int tid = threadIdx.x;
    __shared__ float red[256];

    float m = NEG_INF;
    for (int j = tid; j < NN; j += 256) m = fmaxf(m, p[j]);
    red[tid] = m;
    __syncthreads();
    for (int s = 128; s > 0; s >>= 1) {
        if (tid < s) red[tid] = fmaxf(red[tid], red[tid + s]);
        __syncthreads();
    }
    m = red[0];
    __syncthreads();

    float sum = 0.f;
    for (int j = tid; j < NN; j += 256) {
        const float v = p[j];
        const float ev = (v == NEG_INF) ? 0.f : __expf(v - m);
        p[j] = ev;
        sum += ev;
    }
    red[tid] = sum;
    __syncthreads();
    for (int s = 128; s > 0; s >>= 1) {
        if (tid < s) red[tid] += red[tid + s];
        __syncthreads();
    }
    const float inv = 1.f / red[0];
    for (int j = tid; j < NN; j += 256) p[j] *= inv;
}

// ---------------------------------------------------------------------------
// Kernel 4: global-branch aggregation out[b,i,d] = leaky(sum_j w[b,i,j]*gr[b,j,d])
// Batched (166x166)@(166x400) GEMM on v_wmma_f32_16x16x4_f32, K padded to 168.
// Guards are cndmask-selects, EXEC stays all-1s through the MMA loop.
// ---------------------------------------------------------------------------
__global__ __launch_bounds__(32) void agg_wmma(
    const float* __restrict__ w, const float* __restrict__ gr,
    float* __restrict__ out) {
    const int b  = blockIdx.z;
    const int m0 = blockIdx.x * 16;
    const int n0 = blockIdx.y * 16;
    const int lane = threadIdx.x;
    const int hi = lane >> 4, l = lane & 15;

    const float* __restrict__ wb = w  + (size_t)b * NN * NN;
    const float* __restrict__ gb = gr + (size_t)b * NN * FF;

    const int rowA = m0 + l;
    const bool rowok = rowA < NN;
    v8f acc = {};
    for (int k = 0; k < 168; k += 4) {
        const int ka = k + hi * 2;
        v2f a, bb;
        a.x  = (rowok && ka     < NN) ? wb[(size_t)rowA * NN + ka]     : 0.f;
        a.y  = (rowok && ka + 1 < NN) ? wb[(size_t)rowA * NN + ka + 1] : 0.f;
        bb.x = (ka     < NN) ? gb[(size_t)(ka    ) * FF + n0 + l] : 0.f;
        bb.y = (ka + 1 < NN) ? gb[(size_t)(ka + 1) * FF + n0 + l] : 0.f;
        acc = __builtin_amdgcn_wmma_f32_16x16x4_f32(
            false, a, false, bb, (short)0, acc, false, false);
    }
#pragma unroll
    for (int v = 0; v < 8; ++v) {
        const int row = m0 + v + hi * 8;
        if (row < NN)
            out[((size_t)b * NN + row) * FF + n0 + l] = leaky(acc[v]);
    }
}

// ---------------------------------------------------------------------------
// Kernel 5: local branch. Per (b,i) row: top-5 over j of e_l, softmax over those
// 5, sparse aggregate of gr_l, leaky, accumulate into out (out_g already there).
// ---------------------------------------------------------------------------
__global__ __launch_bounds__(64) void local_topk_agg(
    const float* __restrict__ el, const float* __restrict__ grl,
    float* __restrict__ out) {
    const int row = blockIdx.x;           // b*N + i
    const int b = row / NN;
    const float* __restrict__ p = el + (size_t)row * NN;

    __shared__ float s_w[TOPM];
    __shared__ int   s_idx[TOPM];

    if (threadIdx.x == 0) {
        float tv[TOPM];
        int   ti[TOPM];
#pragma unroll
        for (int m = 0; m < TOPM; ++m) { tv[m] = NEG_INF; ti[m] = -1; }
        for (int j = 0; j < NN; ++j) {
            const float v = p[j];
            if (v > tv[TOPM - 1]) {
                int m = TOPM - 1;
                while (m > 0 && v > tv[m - 1]) {
                    tv[m] = tv[m - 1]; ti[m] = ti[m - 1]; --m;
                }
                tv[m] = v; ti[m] = j;
            }
        }
        const float mx = tv[0];
        float s = 0.f, wv[TOPM];
#pragma unroll
        for (int m = 0; m < TOPM; ++m) {
            wv[m] = (tv[m] == NEG_INF) ? 0.f : __expf(tv[m] - mx);
            s += wv[m];
        }
        const float inv = 1.f / s;
#pragma unroll
        for (int m = 0; m < TOPM; ++m) {
            s_w[m] = wv[m] * inv;
            s_idx[m] = ti[m] < 0 ? 0 : ti[m];
        }
    }
    __syncthreads();

    const float* __restrict__ gb = grl + (size_t)b * NN * FF;
    const float w0 = s_w[0], w1 = s_w[1], w2 = s_w[2], w3 = s_w[3], w4 = s_w[4];
    const float* __restrict__ r0 = gb + (size_t)s_idx[0] * FF;
    const float* __restrict__ r1 = gb + (size_t)s_idx[1] * FF;
    const float* __restrict__ r2 = gb + (size_t)s_idx[2] * FF;
    const float* __restrict__ r3 = gb + (size_t)s_idx[3] * FF;
    const float* __restrict__ r4 = gb + (size_t)s_idx[4] * FF;
    for (int d = threadIdx.x; d < FF; d += 64) {
        float x = w0 * r0[d] + w1 * r1[d] + w2 * r2[d] + w3 * r3[d] + w4 * r4[d];
        out[(size_t)row * FF + d] += leaky(x);
    }
}

// ---------------------------------------------------------------------------
extern "C" void kernel_launch(void* const* d_in, const int* in_sizes, int n_in,
                              void* d_out, int out_size, void* d_ws, size_t ws_size,
                              hipStream_t stream) {
    const float* h    = (const float*)d_in[0];
    const float* adj  = (const float*)d_in[1];
    const float* Wl_g = (const float*)d_in[2];
    const float* Wr_g = (const float*)d_in[3];
    const float* a_g  = (const float*)d_in[4];
    const float* Wl_l = (const float*)d_in[5];
    const float* Wr_l = (const float*)d_in[6];
    const float* a_l  = (const float*)d_in[7];
    const float* b_l  = (const float*)d_in[8];
    float* out = (float*)d_out;

    // Workspace layout (floats): 4 projections then two score matrices (~20.5 MB)
    float* ws = (float*)d_ws;
    const size_t gsz = (size_t)MROWS * DHH;       // 1,062,400
    float* gl_g = ws;                  // z = 0
    float* gr_g = gl_g + gsz;          // z = 1
    float* gl_l = gr_g + gsz;          // z = 2
    float* gr_l = gl_l + gsz;          // z = 3
    float* e_g  = gr_l + gsz;
    float* e_l  = e_g + (size_t)BB * NN * NN;

    // 1) four projections fused, WMMA f32
    proj_wmma<<<dim3(MROWS / 16, DHH / 16), 32, 0, stream>>>(
        h, Wl_g, Wr_g, Wl_l, Wr_l, gl_g);

    // 2) pairwise scores, both branches (TDM-staged gr tiles)
    dim3 sg((NN + 15) / 16, (NN + 15) / 16, BB);
    scores_kernel<<<sg, 256, 0, stream>>>(gl_g, gr_g, a_g, adj, b_l, 0, e_g);
    scores_kernel<<<sg, 256, 0, stream>>>(gl_l, gr_l, a_l, adj, b_l, 1, e_l);

    // 3) global-branch softmax over j
    softmax_rows<<<BB * NN, 256, 0, stream>>>(e_g);

    // 4) global-branch aggregation (WMMA f32) -> writes leaky(out_g)
    agg_wmma<<<dim3((NN + 15) / 16, DHH / 16, BB), 32, 0, stream>>>(e_g, gr_g, out);

    // 5) local branch: top-5 softmax + sparse aggregate, accumulate into out
    local_topk_agg<<<BB * NN, 64, 0, stream>>>(e_l, gr_l, out);
}